// OccGridAccelBatched_Base_21242908246557
// MI455X (gfx1250) — compile-verified
//
#include <hip/hip_runtime.h>

typedef __attribute__((ext_vector_type(2))) float v2f;
typedef __attribute__((ext_vector_type(4))) float v4f;
typedef __attribute__((ext_vector_type(8))) float v8f;

#define STEP_SIZE 0.005f
#define GRES 128

// One wave32 owns 16 rays. V_WMMA_F32_16X16X4_F32 computes a 16(step)x16(ray)
// tile of t / pts.x / pts.y / pts.z per chunk:
//   A (16x4, K=2 used): A[m][0] = (m+0.5)*dt, A[m][1] = 1      (constant)
//   B (4x16):           B[0][r] = coef_r,     B[1][r] = const_r (per chunk)
// A layout (32-bit 16x4): lanes 0-15 hold K=0,1 (M=lane); lanes 16-31 hold K=2,3 (zero).
// B layout (4x16, 2 VGPRs): lanes 0-15 hold rows 0,1 (N=lane); lanes 16-31 rows 2,3 (zero).
// D layout (16x16 f32): VGPR v, lanes 0-15 -> M=v (steps 0-7), lanes 16-31 -> M=v+8.
// => lane L holds 8 CONSECUTIVE steps of ray (L&15): perfect b128 store shape.

__global__ __launch_bounds__(256) void occgrid_march_wmma(
    const unsigned char* __restrict__ occ,     // [4,128,128,128] bool (uniform base)
    const float*         __restrict__ rays_o,  // [N,3]
    const float*         __restrict__ rays_d,  // [N,3]
    const int*           __restrict__ rays_bidx,
    const int*           __restrict__ max_steps_p,
    float*               __restrict__ out,     // pts | t | mask, concatenated
    int N)
{
    const int S      = *max_steps_p;           // 512
    const int lane   = threadIdx.x & 31;
    const int wave   = threadIdx.x >> 5;
    const int rlocal = lane & 15;
    const int half   = lane >> 4;

    int  r     = (blockIdx.x * 8 + wave) * 16 + rlocal;
    bool valid = r < N;
    int  rc    = valid ? r : (N - 1);

    // ---- per-ray setup (both lane-halves compute the same ray's state) ----
    float ox = rays_o[3*rc+0], oy = rays_o[3*rc+1], oz = rays_o[3*rc+2];
    float dx = rays_d[3*rc+0], dy = rays_d[3*rc+1], dz = rays_d[3*rc+2];

    const float eps = 1e-10f;
    float sdx = (fabsf(dx) < eps) ? ((dx >= 0.f) ? eps : -eps) : dx;
    float sdy = (fabsf(dy) < eps) ? ((dy >= 0.f) ? eps : -eps) : dy;
    float sdz = (fabsf(dz) < eps) ? ((dz >= 0.f) ? eps : -eps) : dz;
    float ix = 1.f / sdx, iy = 1.f / sdy, iz = 1.f / sdz;
    float t0x = (-1.f - ox) * ix, t1x = (1.f - ox) * ix;
    float t0y = (-1.f - oy) * iy, t1y = (1.f - oy) * iy;
    float t0z = (-1.f - oz) * iz, t1z = (1.f - oz) * iz;
    float tmin = fmaxf(fmaxf(fminf(t0x,t1x), fminf(t0y,t1y)), fminf(t0z,t1z));
    float tmax = fminf(fminf(fmaxf(t0x,t1x), fmaxf(t0y,t1y)), fmaxf(t0z,t1z));
    float nearr = fmaxf(tmin, 0.f);
    bool  boxok = tmax > nearr;

    // 32-bit voxel offset: batch base (bidx * 128^3 bytes) fits in 23 bits.
    // Keeping `occ` as the uniform SGPR base + per-lane u32 offset lets the
    // backend use GVS addressing (global_load_u8 v, voff, s[base]) instead of
    // a per-sample 64-bit VGPR address add.
    const unsigned int obase = ((unsigned int)rays_bidx[rc]) << 21;

    // ---- constant A operand ----
    float w = (half == 0) ? 1.f : 0.f;             // zero K=2,3 rows
    v2f a;
    a.x = w * (((float)rlocal) + 0.5f) * STEP_SIZE; // A[m][0]
    a.y = w;                                        // A[m][1]

    float*       ptsb  = out;
    float*       tb    = out + (size_t)N * S * 3;
    float*       maskb = out + (size_t)N * S * 4;
    const v8f    zeroC = {};

    const int nchunk = S >> 4;                      // S == 512 -> 32 chunks
    for (int c = 0; c < nchunk; ++c) {
        const int   s_base = c << 4;
        const float t_off  = fmaf((float)s_base, STEP_SIZE, nearr);

        v2f b_t; b_t.x = w;        b_t.y = w * t_off;
        v2f b_x; b_x.x = w * dx;   b_x.y = w * fmaf(t_off, dx, ox);
        v2f b_y; b_y.x = w * dy;   b_y.y = w * fmaf(t_off, dy, oy);
        v2f b_z; b_z.x = w * dz;   b_z.y = w * fmaf(t_off, dz, oz);

        v8f Dt = __builtin_amdgcn_wmma_f32_16x16x4_f32(false, a, false, b_t, (short)0, zeroC, false, false);
        v8f Dx = __builtin_amdgcn_wmma_f32_16x16x4_f32(false, a, false, b_x, (short)0, zeroC, false, false);
        v8f Dy = __builtin_amdgcn_wmma_f32_16x16x4_f32(false, a, false, b_y, (short)0, zeroC, false, false);
        v8f Dz = __builtin_amdgcn_wmma_f32_16x16x4_f32(false, a, false, b_z, (short)0, zeroC, false, false);

        // this lane: ray r, steps s0..s0+7 (contiguous)
        float T[8], Mk[8], P[24];
        #pragma unroll
        for (int v = 0; v < 8; ++v) {
            float tv = Dt[v];
            float px = Dx[v], py = Dy[v], pz = Dz[v];
            bool  inb = (tv < tmax) && boxok;
            int gx = (int)(fmaf(px, 0.5f, 0.5f) * (float)GRES);
            int gy = (int)(fmaf(py, 0.5f, 0.5f) * (float)GRES);
            int gz = (int)(fmaf(pz, 0.5f, 0.5f) * (float)GRES);
            gx = gx < 0 ? 0 : (gx > GRES-1 ? GRES-1 : gx);
            gy = gy < 0 ? 0 : (gy > GRES-1 ? GRES-1 : gy);
            gz = gz < 0 ? 0 : (gz > GRES-1 ? GRES-1 : gz);
            unsigned int voff = obase | ((unsigned int)gx << 14)
                                      | ((unsigned int)gy << 7)
                                      |  (unsigned int)gz;
            unsigned char ob = occ[voff];
            float m = (inb && ob) ? 1.f : 0.f;
            T[v]  = tv;
            Mk[v] = m;
            P[3*v+0] = px * m;
            P[3*v+1] = py * m;
            P[3*v+2] = pz * m;
        }

        if (valid) {
            const int    s0  = s_base + half * 8;
            const size_t row = (size_t)r * S + s0;        // multiple of 8
            float* tp = tb    + row;
            float* mp = maskb + row;
            float* pp = ptsb  + row * 3;                  // multiple of 24
            #pragma unroll
            for (int k = 0; k < 2; ++k) {
                v4f tv4 = { T[4*k],  T[4*k+1],  T[4*k+2],  T[4*k+3]  };
                v4f mv4 = { Mk[4*k], Mk[4*k+1], Mk[4*k+2], Mk[4*k+3] };
                __builtin_nontemporal_store(tv4, (v4f*)(tp + 4*k));
                __builtin_nontemporal_store(mv4, (v4f*)(mp + 4*k));
            }
            #pragma unroll
            for (int k = 0; k < 6; ++k) {
                v4f pv4 = { P[4*k], P[4*k+1], P[4*k+2], P[4*k+3] };
                __builtin_nontemporal_store(pv4, (v4f*)(pp + 4*k));
            }
        }
    }
}

extern "C" void kernel_launch(void* const* d_in, const int* in_sizes, int n_in,
                              void* d_out, int out_size, void* d_ws, size_t ws_size,
                              hipStream_t stream) {
    const unsigned char* occ    = (const unsigned char*)d_in[0];
    const float*         rays_o = (const float*)d_in[1];
    const float*         rays_d = (const float*)d_in[2];
    const int*           bidx   = (const int*)d_in[3];
    const int*           msteps = (const int*)d_in[4];
    float*               out    = (float*)d_out;

    const int N = in_sizes[1] / 3;                 // 32768
    const int raysPerBlock = 128;                  // 8 waves x 16 rays
    const int blocks = (N + raysPerBlock - 1) / raysPerBlock;
    occgrid_march_wmma<<<blocks, 256, 0, stream>>>(occ, rays_o, rays_d, bidx,
                                                   msteps, out, N);
}